// NTMMemory_77318001263251
// MI455X (gfx1250) — compile-verified
//
#include <hip/hip_runtime.h>
#include <math.h>

#define BB 128
#define HH 4
#define SS 1024
#define MM 256

typedef __attribute__((ext_vector_type(2))) float v2f;
typedef __attribute__((ext_vector_type(8))) float v8f;

// ---------------------------------------------------------------------------
// Kernel 1: fused  dots[b,s,h] = k[b,h,:]·memory[b,s,:]  and  mn[b,s]=||row||
// One wave handles 16 s-rows; D(16x16) = A(16 rows x K4) * B(K4 x 16 heads)
// via V_WMMA_F32_16X16X4_F32, K=256 walked in 64 chunks. Heads padded 4->16
// with zero rows in LDS so every lane loads unconditionally (EXEC all ones).
// ---------------------------------------------------------------------------
__global__ __launch_bounds__(256) void k_dots(const float* __restrict__ mem,
                                              const float* __restrict__ kk,
                                              float* __restrict__ dots,
                                              float* __restrict__ mn) {
  __shared__ float kls[16 * MM];  // 16 "heads" x 256, rows 4..15 zero
  const int b = blockIdx.y;
  const int chunk = blockIdx.x;
  const int tid = threadIdx.x;
  for (int i = tid; i < 16 * MM; i += 256)
    kls[i] = (i < HH * MM) ? kk[(size_t)b * HH * MM + i] : 0.f;
  __syncthreads();

  const int lane = tid & 31;
  const int wave = tid >> 5;
  const int srow = lane & 15;   // A-matrix row for this lane
  const int kh   = lane >> 4;   // which K-half (K0/K1 vs K2/K3)
  const int h    = lane & 15;   // B/D column (head, padded)
  const int s_base = chunk * 128 + wave * 16;

  const float* mrow = mem + ((size_t)b * SS + s_base + srow) * MM;
  v8f c = {0.f, 0.f, 0.f, 0.f, 0.f, 0.f, 0.f, 0.f};
  float ss = 0.f;

  for (int kc = 0; kc < MM; kc += 4) {
    const int ko = kc + 2 * kh;
    const v2f a  = *(const v2f*)(mrow + ko);           // A: 16x4 fp32 layout
    const v2f bf = *(const v2f*)(kls + h * MM + ko);   // B: 4x16 fp32 layout
    ss += a.x * a.x + a.y * a.y;                       // fused row-norm
#if __has_builtin(__builtin_amdgcn_wmma_f32_16x16x4_f32)
    c = __builtin_amdgcn_wmma_f32_16x16x4_f32(false, a, false, bf,
                                              (short)0, c, false, false);
#else
    c[0] += a.x * bf.x + a.y * bf.y;  // placeholder fallback (not expected)
#endif
  }

  // row sumsq: combine the two K-halves (lanes l and l+16 hold same row)
  const float tot = ss + __shfl_xor(ss, 16, 32);
  if (lane < 16) mn[(size_t)b * SS + s_base + lane] = sqrtf(tot);

  float cc[8];
#pragma unroll
  for (int v = 0; v < 8; ++v) cc[v] = c[v];
  if (h < HH) {
#pragma unroll
    for (int v = 0; v < 8; ++v) {
      const int sr = v + 8 * kh;  // C/D layout: lanes<16 -> M=v, >=16 -> M=v+8
      dots[((size_t)b * SS + s_base + sr) * HH + h] = cc[v];
    }
  }
}

// ---------------------------------------------------------------------------
// Block-wide reductions (1024 threads = 32 waves exactly)
// ---------------------------------------------------------------------------
__device__ __forceinline__ float block_max(float v, float* red) {
#pragma unroll
  for (int o = 16; o > 0; o >>= 1) v = fmaxf(v, __shfl_xor(v, o, 32));
  const int lane = threadIdx.x & 31, w = threadIdx.x >> 5;
  if (lane == 0) red[w] = v;
  __syncthreads();
  if (w == 0) {
    float x = red[lane];
#pragma unroll
    for (int o = 16; o > 0; o >>= 1) x = fmaxf(x, __shfl_xor(x, o, 32));
    if (lane == 0) red[0] = x;
  }
  __syncthreads();
  const float r = red[0];
  __syncthreads();
  return r;
}

__device__ __forceinline__ float block_sum(float v, float* red) {
#pragma unroll
  for (int o = 16; o > 0; o >>= 1) v += __shfl_xor(v, o, 32);
  const int lane = threadIdx.x & 31, w = threadIdx.x >> 5;
  if (lane == 0) red[w] = v;
  __syncthreads();
  if (w == 0) {
    float x = red[lane];
#pragma unroll
    for (int o = 16; o > 0; o >>= 1) x += __shfl_xor(x, o, 32);
    if (lane == 0) red[0] = x;
  }
  __syncthreads();
  const float r = red[0];
  __syncthreads();
  return r;
}

// ---------------------------------------------------------------------------
// Kernel 2: one block per batch; thread = s. softmax(beta*cos), interpolate,
// circular shift (+1,0,-1 with weights s0,s1,s2), sharpen, renormalize -> w.
// ---------------------------------------------------------------------------
__global__ __launch_bounds__(1024) void k_weights(
    const float* __restrict__ kk, const float* __restrict__ beta,
    const float* __restrict__ pw, const float* __restrict__ g,
    const float* __restrict__ sw, const float* __restrict__ gamma,
    const float* __restrict__ dots, const float* __restrict__ mn,
    float* __restrict__ w) {
  __shared__ float red[32];
  __shared__ float knl[HH];
  __shared__ float wgl[SS];
  const int b = blockIdx.x;
  const int s = threadIdx.x;

  if (s < HH) {  // key norms (tiny)
    const float* kr = kk + ((size_t)b * HH + s) * MM;
    float t = 0.f;
    for (int m = 0; m < MM; ++m) t += kr[m] * kr[m];
    knl[s] = sqrtf(t);
  }
  __syncthreads();

  const float mnv = mn[(size_t)b * SS + s];
  for (int h = 0; h < HH; ++h) {
    const float dot  = dots[((size_t)b * SS + s) * HH + h];
    const float cosv = dot / fmaxf(knl[h] * mnv, 1e-8f);
    const float x    = beta[b * HH + h] * cosv;
    const float xm   = block_max(x, red);
    const float p    = expf(x - xm);
    const float psum = block_sum(p, red);
    const float wc   = p / psum;
    const float gv   = g[b * HH + h];
    const float wg   = gv * wc + (1.f - gv) * pw[((size_t)b * HH + h) * SS + s];
    wgl[s] = wg;
    __syncthreads();
    const float s0 = sw[(b * HH + h) * 3 + 0];
    const float s1 = sw[(b * HH + h) * 3 + 1];
    const float s2 = sw[(b * HH + h) * 3 + 2];
    // ws[i] = wg[i-1]*s0 + wg[i]*s1 + wg[i+1]*s2  (circular)
    const float wsv = wgl[(s + SS - 1) & (SS - 1)] * s0 + wg * s1 +
                      wgl[(s + 1) & (SS - 1)] * s2;
    __syncthreads();
    const float wgam = powf(wsv, gamma[b * HH + h]);
    const float tsum = block_sum(wgam, red);
    w[((size_t)b * HH + h) * SS + s] = wgam / tsum;
  }
}

// ---------------------------------------------------------------------------
// Kernel 3: nmem = memory*(1 - wl*erase_l) + wl*add_l  (head H-1), fused with
// per-chunk partial read vectors  r_part[h,m] += w[h,s]*nmem[s,m].
// Thread = column m (fully coalesced); 128 rows per block. nmem stored
// non-temporally so the `memory` tensor stays resident in the 192MB L2.
// ---------------------------------------------------------------------------
__global__ __launch_bounds__(256) void k_update(
    const float* __restrict__ mem, const float* __restrict__ erase,
    const float* __restrict__ add, const float* __restrict__ w,
    float* __restrict__ nmem, float* __restrict__ partial) {
  __shared__ float wld[HH][128];
  const int b = blockIdx.y, chunk = blockIdx.x;
  const int m = threadIdx.x;
  for (int i = m; i < HH * 128; i += 256) {
    const int h = i >> 7, r = i & 127;
    wld[h][r] = w[((size_t)b * HH + h) * SS + chunk * 128 + r];
  }
  __syncthreads();

  const float er = erase[((size_t)b * HH + (HH - 1)) * MM + m];
  const float ad = add  [((size_t)b * HH + (HH - 1)) * MM + m];
  const float* mrow = mem  + ((size_t)b * SS + chunk * 128) * MM;
  float*       orow = nmem + ((size_t)b * SS + chunk * 128) * MM;

  float a0 = 0.f, a1 = 0.f, a2 = 0.f, a3 = 0.f;
  for (int r = 0; r < 128; ++r) {
    const float wl = wld[HH - 1][r];
    const float nm = mrow[(size_t)r * MM + m] * (1.f - wl * er) + wl * ad;
    __builtin_nontemporal_store(nm, orow + (size_t)r * MM + m);
    a0 += wld[0][r] * nm;
    a1 += wld[1][r] * nm;
    a2 += wld[2][r] * nm;
    a3 += wld[3][r] * nm;
  }
  float* pp = partial + (((size_t)b * 8 + chunk) * HH) * MM + m;
  pp[0 * MM] = a0; pp[1 * MM] = a1; pp[2 * MM] = a2; pp[3 * MM] = a3;
}

// ---------------------------------------------------------------------------
// Kernel 4: deterministic reduction of the 8 s-chunk partials -> r[b,h,m]
// ---------------------------------------------------------------------------
__global__ __launch_bounds__(256) void k_reduce_r(
    const float* __restrict__ partial, float* __restrict__ r) {
  const int idx = blockIdx.x * 256 + threadIdx.x;  // over B*H*M
  const int m  = idx & (MM - 1);
  const int bh = idx >> 8;
  const int h  = bh & (HH - 1);
  const int b  = bh >> 2;
  float t = 0.f;
#pragma unroll
  for (int c = 0; c < 8; ++c)
    t += partial[(((size_t)b * 8 + c) * HH + h) * MM + m];
  r[idx] = t;
}

extern "C" void kernel_launch(void* const* d_in, const int* in_sizes, int n_in,
                              void* d_out, int out_size, void* d_ws,
                              size_t ws_size, hipStream_t stream) {
  const float* mem   = (const float*)d_in[0];
  const float* kk    = (const float*)d_in[1];
  const float* beta  = (const float*)d_in[2];
  const float* pw    = (const float*)d_in[3];
  const float* g     = (const float*)d_in[4];
  const float* sw    = (const float*)d_in[5];
  const float* gamma = (const float*)d_in[6];
  const float* erase = (const float*)d_in[7];
  const float* add   = (const float*)d_in[8];

  float* out  = (float*)d_out;
  float* nmem = out;                          // (B,S,M)
  float* r    = out + (size_t)BB * SS * MM;   // (B,H,M)

  float* dots    = (float*)d_ws;                     // B*S*H
  float* mn      = dots + (size_t)BB * SS * HH;      // B*S
  float* w       = mn   + (size_t)BB * SS;           // B*H*S
  float* partial = w    + (size_t)BB * HH * SS;      // B*8*H*M

  k_dots    <<<dim3(SS / 128, BB), 256,  0, stream>>>(mem, kk, dots, mn);
  k_weights <<<dim3(BB),           1024, 0, stream>>>(kk, beta, pw, g, sw,
                                                      gamma, dots, mn, w);
  k_update  <<<dim3(SS / 128, BB), 256,  0, stream>>>(mem, erase, add, w,
                                                      nmem, partial);
  k_reduce_r<<<dim3((BB * HH * MM) / 256), 256, 0, stream>>>(partial, r);
}